// MLPDecoder_88476326297882
// MI455X (gfx1250) — compile-verified
//
#include <hip/hip_runtime.h>

typedef float v2f __attribute__((ext_vector_type(2)));
typedef float v8f __attribute__((ext_vector_type(8)));

#define DIM 128

// One wave32 computes sigmoid(|X[r]-X[c]| . w) for a tile of 16 edges.
// Lanes 0-15  : M = edge 0..15 of tile, stream dims   0..63 of their rows.
// Lanes 16-31 : M = edge 0..15 of tile, stream dims  64..127 of their rows.
// K-reduction done by chained V_WMMA_F32_16X16X4_F32 (full fp32 precision).
__global__ __launch_bounds__(256)
void mlp_decoder_wmma_kernel(const float* __restrict__ X,
                             const float* __restrict__ W,
                             const int*   __restrict__ R,
                             const int*   __restrict__ C,
                             float*       __restrict__ out,
                             int n_edges) {
    const int lane      = threadIdx.x & 31;
    const int wave_id   = blockIdx.x * (blockDim.x >> 5) + (threadIdx.x >> 5);
    const int edge_base = wave_id * 16;
    const int m         = lane & 15;        // edge-in-tile (matrix row M)
    const bool hi       = lane >= 16;       // high half-wave -> dims 64..127
    const int half_off  = hi ? 64 : 0;

    // Clamp tail edges so all 32 lanes stay active (WMMA needs EXEC all-1s).
    int e = edge_base + m;
    if (e >= n_edges) e = n_edges - 1;

    const float4* __restrict__ a4 =
        reinterpret_cast<const float4*>(X + (size_t)R[e] * DIM + half_off);
    const float4* __restrict__ b4 =
        reinterpret_cast<const float4*>(X + (size_t)C[e] * DIM + half_off);
    const float4* __restrict__ w4 =
        reinterpret_cast<const float4*>(W + half_off);

    v8f acc = {};   // 16x16 f32 C/D tile (all 16 N-columns identical by construction)

    #pragma unroll
    for (int u = 0; u < 16; ++u) {          // 16 float4 chunks cover this lane's 64 dims
        float4 a = a4[u];
        float4 b = b4[u];
        float4 w = w4[u];

        // A-matrix slots: lanes<16 give K0,K1 ; lanes>=16 give K2,K3
        v2f A0 = { __builtin_fabsf(a.x - b.x), __builtin_fabsf(a.y - b.y) };
        v2f A1 = { __builtin_fabsf(a.z - b.z), __builtin_fabsf(a.w - b.w) };
        // B-matrix: same w value broadcast to every N column
        v2f B0 = { w.x, w.y };
        v2f B1 = { w.z, w.w };

        acc = __builtin_amdgcn_wmma_f32_16x16x4_f32(
                  false, A0, false, B0, (short)0, acc, false, false);
        acc = __builtin_amdgcn_wmma_f32_16x16x4_f32(
                  false, A1, false, B1, (short)0, acc, false, false);
    }

    // D layout: VGPR r, lanes 0-15 -> M=r ; lanes 16-31 -> M=r+8.
    // All N columns equal, so lane 0 carries edges base+0..7 and
    // lane 16 carries edges base+8..15 in acc[0..7].
    if (m == 0) {
        const int base = edge_base + (hi ? 8 : 0);
        #pragma unroll
        for (int r = 0; r < 8; ++r) {
            const int eo = base + r;
            if (eo < n_edges) {
                const float logit = acc[r];
                out[eo] = 1.0f / (1.0f + __expf(-logit));
            }
        }
    }
}

extern "C" void kernel_launch(void* const* d_in, const int* in_sizes, int n_in,
                              void* d_out, int out_size, void* d_ws, size_t ws_size,
                              hipStream_t stream) {
    const float* X  = (const float*)d_in[0];   // [N_NODES, 128] f32
    const float* W  = (const float*)d_in[1];   // [128, 1] f32
    const int*   R  = (const int*)d_in[2];     // [E] i32
    const int*   C  = (const int*)d_in[3];     // [E] i32
    float*       o  = (float*)d_out;           // [E] f32
    const int n_edges = in_sizes[2];

    // 8 waves/block * 16 edges/wave = 128 edges per block of 256 threads.
    const int edges_per_block = 128;
    const int grid = (n_edges + edges_per_block - 1) / edges_per_block;
    mlp_decoder_wmma_kernel<<<grid, 256, 0, stream>>>(X, W, R, C, o, n_edges);
}